// TemporalGNN_63866163691548
// MI455X (gfx1250) — compile-verified
//
#include <hip/hip_runtime.h>
#include <math.h>

typedef float v2f __attribute__((ext_vector_type(2)));
typedef float v8f __attribute__((ext_vector_type(8)));

#define PER 12
#define FPC 48   // F*P channels per node
#define HID 64

// ---------------------------------------------------------------------------
// Small prep: softmax(att) -> probs; fold GCN weight into Linear top half:
//   Wzeff[f][n] = sum_k Wg_z[f][k] * Wl_z[k][n]   (Wl top half, k<64)
//   cz[n]      = sum_k bg_z[k]*Wl_z[k][n] + bl_z[n]
// ---------------------------------------------------------------------------
__global__ void prep_small_kernel(const float* __restrict__ att,
    const float* __restrict__ Wgz, const float* __restrict__ bgz,
    const float* __restrict__ Wlz, const float* __restrict__ blz,
    const float* __restrict__ Wgh, const float* __restrict__ bgh,
    const float* __restrict__ Wlh, const float* __restrict__ blh,
    float* __restrict__ wz, float* __restrict__ wh,
    float* __restrict__ ce, float* __restrict__ pr) {
  const int n = threadIdx.x;  // 0..63
  if (n == 0) {
    float mx = att[0];
    for (int p = 1; p < PER; ++p) mx = fmaxf(mx, att[p]);
    float e[PER], s = 0.f;
    for (int p = 0; p < PER; ++p) { e[p] = expf(att[p] - mx); s += e[p]; }
    for (int p = 0; p < PER; ++p) pr[p] = e[p] / s;
  }
  float cz = blz[n], ch = blh[n];
  for (int k = 0; k < HID; ++k) {
    cz += bgz[k] * Wlz[k * HID + n];
    ch += bgh[k] * Wlh[k * HID + n];
  }
  ce[n] = cz; ce[HID + n] = ch;
  for (int f = 0; f < 4; ++f) {
    float az = 0.f, ah = 0.f;
    for (int k = 0; k < HID; ++k) {
      az += Wgz[f * HID + k] * Wlz[k * HID + n];
      ah += Wgh[f * HID + k] * Wlh[k * HID + n];
    }
    wz[f * HID + n] = az;
    wh[f * HID + n] = ah;
  }
}

// deg[n] = 1 (self loop), then += edge weights at dst, then rsqrt in place
__global__ void deg_init_kernel(float* __restrict__ deg, int Nn) {
  int n = blockIdx.x * blockDim.x + threadIdx.x;
  if (n < Nn) deg[n] = 1.0f;
}

__global__ void deg_accum_kernel(const int* __restrict__ ei,
                                 const float* __restrict__ ew,
                                 float* __restrict__ deg, int Ee) {
  int e = blockIdx.x * blockDim.x + threadIdx.x;
  if (e < Ee) atomicAdd(&deg[ei[Ee + e]], ew[e]);
}

__global__ void dinv_kernel(float* __restrict__ d, int Nn) {
  int n = blockIdx.x * blockDim.x + threadIdx.x;
  if (n < Nn) {
    float v = d[n];
    d[n] = v > 0.f ? rsqrtf(v) : 0.f;
  }
}

// Propagate raw x (N x 48) through normalized adjacency (real edges only;
// self loops are folded into the node kernel). One thread = one edge x 4 ch.
__global__ void propagate_kernel(const float* __restrict__ x,
                                 const int* __restrict__ ei,
                                 const float* __restrict__ ew,
                                 const float* __restrict__ dinv,
                                 float* __restrict__ agg, int Ee) {
  int tid = blockIdx.x * blockDim.x + threadIdx.x;
  int e = tid / PER;
  if (e >= Ee) return;
  int cg = tid - e * PER;             // 0..11 -> channels cg*4 .. cg*4+3
  int s = ei[e], d = ei[Ee + e];
  float nrm = dinv[s] * ew[e] * dinv[d];
  const float4 xv = *(const float4*)(x + (size_t)s * FPC + cg * 4);
  float* ap = agg + (size_t)d * FPC + cg * 4;
  atomicAdd(ap + 0, nrm * xv.x);
  atomicAdd(ap + 1, nrm * xv.y);
  atomicAdd(ap + 2, nrm * xv.z);
  atomicAdd(ap + 3, nrm * xv.w);
}

// ---------------------------------------------------------------------------
// WMMA f32 16x16x4 dense tile helpers.
// A (16x4 f32): lanes 0-15 hold M=0..15 K=(k0,k0+1); lanes 16-31 K=(k0+2,k0+3)
// B (4x16 f32): vgpr0 = row k0 (lanes 0-15) / row k0+2 (lanes 16-31); vgpr1 +1
// C/D (16x16): vgpr v = row (v + 8*hi), lane&15 = column
// ---------------------------------------------------------------------------
template <int NT>
__device__ __forceinline__ void dense_tile(const float* __restrict__ W,
                                           const float* __restrict__ bias,
                                           const float* __restrict__ in_lds,
                                           int lane, v8f (&c)[NT], int ncols) {
  const int hi = lane >> 4;
  const int nl = lane & 15;
  const int koff = hi ? 2 : 0;
#pragma unroll
  for (int t = 0; t < NT; ++t) {
    float bv = bias[t * 16 + nl];
#pragma unroll
    for (int v = 0; v < 8; ++v) c[t][v] = bv;
  }
#pragma unroll
  for (int ks = 0; ks < 16; ++ks) {
    const int k0 = ks * 4 + koff;
    float2 av = *(const float2*)(in_lds + nl * HID + k0);
    v2f a; a.x = av.x; a.y = av.y;
#pragma unroll
    for (int t = 0; t < NT; ++t) {
      v2f b;
      b.x = W[k0 * ncols + t * 16 + nl];
      b.y = W[(k0 + 1) * ncols + t * 16 + nl];
      c[t] = __builtin_amdgcn_wmma_f32_16x16x4_f32(
          false, a, false, b, (short)0, c[t], false, false);
    }
  }
}

template <int NT>
__device__ __forceinline__ void store_lds_tile(float* __restrict__ out_lds,
                                               int lane, const v8f (&c)[NT],
                                               int ncols) {
  const int hi = lane >> 4;
  const int nl = lane & 15;
#pragma unroll
  for (int t = 0; t < NT; ++t)
#pragma unroll
    for (int v = 0; v < 8; ++v)
      out_lds[(v + hi * 8) * ncols + t * 16 + nl] = c[t][v];
}

// ---------------------------------------------------------------------------
// Fused per-node kernel: self-loop fold + gate stage (VALU) + dense WMMA chain
// One wave (32 lanes) owns one 16-node tile. 4 waves / 128-thread block.
// ---------------------------------------------------------------------------
__global__ __launch_bounds__(128) void fused_node_kernel(
    const float* __restrict__ x, const float* __restrict__ agg,
    const float* __restrict__ dinv,
    const float* __restrict__ wz, const float* __restrict__ wh,
    const float* __restrict__ ce, const float* __restrict__ pr,
    const float* __restrict__ W1, const float* __restrict__ b1,
    const float* __restrict__ W2, const float* __restrict__ b2,
    const float* __restrict__ W3, const float* __restrict__ b3,
    const float* __restrict__ W4, const float* __restrict__ b4,
    const float* __restrict__ Wo, const float* __restrict__ bo,
    float* __restrict__ out, int ntiles) {
  __shared__ float sWz[256], sWh[256], sCz[64], sCh[64], sPr[16];
  __shared__ float sAgg[4][16 * FPC];
  __shared__ float sH[4][2][16 * HID];

  const int tid = threadIdx.x;
  for (int i = tid; i < 256; i += 128) { sWz[i] = wz[i]; sWh[i] = wh[i]; }
  if (tid < 64) { sCz[tid] = ce[tid]; sCh[tid] = ce[64 + tid]; }
  if (tid < PER) sPr[tid] = pr[tid];
  {
    int off = tid * 32;
    if (off < 4096) {
      __builtin_prefetch(W1 + off, 0, 1);
      __builtin_prefetch(W2 + off, 0, 1);
      __builtin_prefetch(W3 + off, 0, 1);
      __builtin_prefetch(W4 + off, 0, 1);
    }
    if (off < 3072) __builtin_prefetch(Wo + off, 0, 1);
  }
  __syncthreads();

  const int wave = tid >> 5;
  const int lane = tid & 31;
  const int tile = blockIdx.x * 4 + wave;
  if (tile >= ntiles) return;   // wave-uniform; EXEC stays all-ones below
  const int base = tile * 16;

  // agg tile + self-loop fold: aggl = A_hat_edges*x + dinv^2 * x(self)
  float* aggl = &sAgg[wave][0];
  for (int i = lane; i < 16 * FPC; i += 32) {
    int m = i / FPC;
    size_t gi = (size_t)(base + m) * FPC + (i - m * FPC);
    float dv = dinv[base + m];
    aggl[i] = agg[gi] + dv * dv * x[gi];
  }

  float* h0 = &sH[wave][0][0];
  float* h1 = &sH[wave][1][0];

  // Stage A: h0[m][n] = relu( sum_p probs[p]*(1-sigmoid(zp))*tanh(hp) )
  for (int idx = lane; idx < 16 * HID; idx += 32) {
    int m = idx >> 6, n = idx & 63;
    const float* ar = aggl + m * FPC;
    float wz0 = sWz[n], wz1 = sWz[64 + n], wz2 = sWz[128 + n], wz3 = sWz[192 + n];
    float wh0 = sWh[n], wh1 = sWh[64 + n], wh2 = sWh[128 + n], wh3 = sWh[192 + n];
    float czv = sCz[n], chv = sCh[n];
    float acc = 0.f;
#pragma unroll
    for (int p = 0; p < PER; ++p) {
      float a0 = ar[p], a1 = ar[12 + p], a2 = ar[24 + p], a3 = ar[36 + p];
      float zp = czv + a0 * wz0 + a1 * wz1 + a2 * wz2 + a3 * wz3;
      float hp = chv + a0 * wh0 + a1 * wh1 + a2 * wh2 + a3 * wh3;
      float zg = 1.f / (1.f + __expf(-zp));
      acc += sPr[p] * (1.f - zg) * tanhf(hp);
    }
    h0[idx] = fmaxf(acc, 0.f);
  }

  // Stage B: dense chain with fp32 WMMA, LDS ping-pong between layers
  v8f c4[4];
  dense_tile<4>(W1, b1, h0, lane, c4, HID); store_lds_tile<4>(h1, lane, c4, HID);
  dense_tile<4>(W2, b2, h1, lane, c4, HID); store_lds_tile<4>(h0, lane, c4, HID);
  dense_tile<4>(W3, b3, h0, lane, c4, HID); store_lds_tile<4>(h1, lane, c4, HID);
  dense_tile<4>(W4, b4, h1, lane, c4, HID); store_lds_tile<4>(h0, lane, c4, HID);

  v8f c3[3];
  dense_tile<3>(Wo, bo, h0, lane, c3, FPC);
  {
    const int hi = lane >> 4, nl = lane & 15;
#pragma unroll
    for (int t = 0; t < 3; ++t)
#pragma unroll
      for (int v = 0; v < 8; ++v)
        out[(size_t)(base + v + hi * 8) * FPC + t * 16 + nl] = c3[t][v];
  }
}

// ---------------------------------------------------------------------------
extern "C" void kernel_launch(void* const* d_in, const int* in_sizes, int n_in,
                              void* d_out, int out_size, void* d_ws,
                              size_t ws_size, hipStream_t stream) {
  const float* x    = (const float*)d_in[0];
  const int*   ei   = (const int*)d_in[1];
  const float* ew   = (const float*)d_in[2];
  const float* att  = (const float*)d_in[3];
  const float* Wgz  = (const float*)d_in[4];
  const float* bgz  = (const float*)d_in[5];
  const float* Wlz  = (const float*)d_in[6];
  const float* blz  = (const float*)d_in[7];
  // indices 8..11 (r gate) are mathematically dead: H starts at zero so Hz*r==0
  const float* Wgh  = (const float*)d_in[12];
  const float* bgh  = (const float*)d_in[13];
  const float* Wlh  = (const float*)d_in[14];
  const float* blh  = (const float*)d_in[15];
  const float* W1 = (const float*)d_in[16]; const float* b1 = (const float*)d_in[17];
  const float* W2 = (const float*)d_in[18]; const float* b2 = (const float*)d_in[19];
  const float* W3 = (const float*)d_in[20]; const float* b3 = (const float*)d_in[21];
  const float* W4 = (const float*)d_in[22]; const float* b4 = (const float*)d_in[23];
  const float* Wo = (const float*)d_in[24]; const float* bo = (const float*)d_in[25];
  float* out = (float*)d_out;

  const int Nn = in_sizes[0] / FPC;   // 50000
  const int Ee = in_sizes[1] / 2;     // 1.6M

  // workspace layout (floats)
  float* agg = (float*)d_ws;                    // Nn * 48
  float* dv  = agg + (size_t)Nn * FPC;          // Nn   (deg -> dinv in place)
  float* wz  = dv + Nn;                         // 256
  float* wh  = wz + 256;                        // 256
  float* ce  = wh + 256;                        // 128
  float* pr  = ce + 128;                        // 16

  hipMemsetAsync(agg, 0, (size_t)Nn * FPC * sizeof(float), stream);

  prep_small_kernel<<<1, 64, 0, stream>>>(att, Wgz, bgz, Wlz, blz,
                                          Wgh, bgh, Wlh, blh, wz, wh, ce, pr);
  deg_init_kernel<<<(Nn + 255) / 256, 256, 0, stream>>>(dv, Nn);
  deg_accum_kernel<<<(Ee + 255) / 256, 256, 0, stream>>>(ei, ew, dv, Ee);
  dinv_kernel<<<(Nn + 255) / 256, 256, 0, stream>>>(dv, Nn);

  long long pthreads = (long long)Ee * PER;
  int pblocks = (int)((pthreads + 255) / 256);
  propagate_kernel<<<pblocks, 256, 0, stream>>>(x, ei, ew, dv, agg, Ee);

  int ntiles = Nn / 16;
  fused_node_kernel<<<(ntiles + 3) / 4, 128, 0, stream>>>(
      x, agg, dv, wz, wh, ce, pr,
      W1, b1, W2, b2, W3, b3, W4, b4, Wo, bo, out, ntiles);
}